// sts_attention_40819369181201
// MI455X (gfx1250) — compile-verified
//
#include <hip/hip_runtime.h>
#include <math.h>

#define HN 3
#define BB 64
#define CN 64
#define TN 256
#define VN 25
#define VP 32
#define TT 8      // time tile, attention main kernel
#define XPS (TT * 28)   // xp/v row-group stride per channel: t*28+w, w<28
#define TT4 16    // time tile, feature extraction kernel
#define HALO 4

typedef float v2f __attribute__((ext_vector_type(2)));
typedef float v8f __attribute__((ext_vector_type(8)));
typedef unsigned int v4u __attribute__((ext_vector_type(4)));
typedef int v4i __attribute__((ext_vector_type(4)));
typedef int v8i __attribute__((ext_vector_type(8)));

#ifndef __has_builtin
#define __has_builtin(x) 0
#endif
#if __has_builtin(__builtin_amdgcn_tensor_load_to_lds) && \
    __has_builtin(__builtin_amdgcn_s_wait_tensorcnt)
#define USE_TDM 1
#else
#define USE_TDM 0
#endif

__device__ __forceinline__ v8f wmma4(v2f a, v2f b, v8f c) {
  // D = A(16x4 f32) * B(4x16 f32) + C(16x16 f32)
  return __builtin_amdgcn_wmma_f32_16x16x4_f32(false, a, false, b, (short)0, c,
                                               false, false);
}

__device__ __forceinline__ v8f vzero8() {
  v8f z;
  for (int i = 0; i < 8; ++i) z[i] = 0.f;
  return z;
}

// ---------------------------------------------------------------------------
// Kernel 1: Sinkhorn on the 3 25x25 log_alpha matrices -> padded P[3][32][32]
// ---------------------------------------------------------------------------
__global__ void k_sinkhorn(const float* __restrict__ log_alpha,
                           float* __restrict__ Ppad) {
  __shared__ float la[VN][VN];
  const int tid = threadIdx.x;  // 32 threads
  for (int h = 0; h < HN; ++h) {
    for (int idx = tid; idx < VN * VN; idx += 32)
      la[idx / VN][idx % VN] = log_alpha[h * VN * VN + idx];
    __syncthreads();
    for (int it = 0; it < 20; ++it) {
      if (tid < VN) {  // rows (axis=1 logsumexp)
        float m = -INFINITY;
        for (int j = 0; j < VN; ++j) m = fmaxf(m, la[tid][j]);
        float s = 0.f;
        for (int j = 0; j < VN; ++j) s += expf(la[tid][j] - m);
        float l = m + logf(s);
        for (int j = 0; j < VN; ++j) la[tid][j] -= l;
      }
      __syncthreads();
      if (tid < VN) {  // cols (axis=0 logsumexp)
        float m = -INFINITY;
        for (int i = 0; i < VN; ++i) m = fmaxf(m, la[i][tid]);
        float s = 0.f;
        for (int i = 0; i < VN; ++i) s += expf(la[i][tid] - m);
        float l = m + logf(s);
        for (int i = 0; i < VN; ++i) la[i][tid] -= l;
      }
      __syncthreads();
    }
    for (int idx = tid; idx < VP * VP; idx += 32) {
      int u = idx >> 5, w = idx & 31;
      Ppad[h * VP * VP + idx] = (u < VN && w < VN) ? expf(la[u][w]) : 0.f;
    }
    __syncthreads();
  }
}

// ---------------------------------------------------------------------------
// Kernel 2: xbar[b][c][u] = mean_T x  (padded u->32 with zeros)
// ---------------------------------------------------------------------------
__global__ void k_xbar(const float* __restrict__ x, float* __restrict__ xbar) {
  int idx = blockIdx.x * blockDim.x + threadIdx.x;
  if (idx >= BB * CN * VP) return;
  int u = idx & 31, bc = idx >> 5;
  float acc = 0.f;
  if (u < VN) {
    const float* p = x + (size_t)bc * TN * VN + u;
    for (int t = 0; t < TN; ++t) acc += p[t * VN];
    acc *= (1.0f / TN);
  }
  xbar[idx] = acc;
}

// ---------------------------------------------------------------------------
// Kernel 3: per (b,h) attention logits -> s_pad[b][h][c][32][32] (zero padded)
// ---------------------------------------------------------------------------
__global__ void k_attn_s(const float* __restrict__ xbar, const float* __restrict__ Ppad,
                         const float* __restrict__ wk, const float* __restrict__ bk,
                         const float* __restrict__ wq, const float* __restrict__ bq,
                         const float* __restrict__ dpw1, const float* __restrict__ dpb1,
                         const float* __restrict__ dpw2, const float* __restrict__ dpb2,
                         const float* __restrict__ ffnw, const float* __restrict__ ffnb,
                         const float* __restrict__ alpha, const float* __restrict__ adjc,
                         float* __restrict__ s_pad) {
  const int h = blockIdx.x, b = blockIdx.y;
  __shared__ float xm[CN][VN];
  __shared__ float kk[4][VN], qq[4][VN];
  __shared__ float aa[8][VN * VN];

  // xm = xbar[b] @ P[h]
  for (int idx = threadIdx.x; idx < CN * VN; idx += blockDim.x) {
    int c = idx / VN, w = idx % VN;
    const float* xb = xbar + (size_t)(b * CN + c) * VP;
    const float* Ph = Ppad + h * VP * VP;
    float acc = 0.f;
    for (int u = 0; u < VN; ++u) acc += xb[u] * Ph[u * VP + w];
    xm[c][w] = acc;
  }
  __syncthreads();
  // k/q projections [4][25]
  for (int idx = threadIdx.x; idx < 2 * 4 * VN; idx += blockDim.x) {
    int which = idx / (4 * VN), r = (idx / VN) & 3, w = idx % VN;
    const float* W = (which == 0 ? wk : wq) + (h * 4 + r) * CN;
    float acc = (which == 0 ? bk : bq)[h * 4 + r];
    for (int c = 0; c < CN; ++c) acc += W[c] * xm[c][w];
    if (which == 0) kk[r][w] = acc; else qq[r][w] = acc;
  }
  __syncthreads();
  // a = tanh([dp1(k*q); dp2(k-q)])  [8][25][25]
  for (int idx = threadIdx.x; idx < VN * VN; idx += blockDim.x) {
    int vv = idx / VN, w = idx % VN;
    float s1[4], s2[4];
    for (int r = 0; r < 4; ++r) { s1[r] = kk[r][vv] * qq[r][w]; s2[r] = kk[r][vv] - qq[r][w]; }
    for (int o = 0; o < 4; ++o) {
      float y1 = dpb1[h * 4 + o], y2 = dpb2[h * 4 + o];
      for (int r = 0; r < 4; ++r) {
        y1 += dpw1[(h * 4 + o) * 4 + r] * s1[r];
        y2 += dpw2[(h * 4 + o) * 4 + r] * s2[r];
      }
      aa[o][idx] = tanhf(y1);
      aa[4 + o][idx] = tanhf(y2);
    }
  }
  __syncthreads();
  // s = alpha*ffn(a) + adjc  -> zero-padded [C][32][32]
  float* sb = s_pad + (size_t)((b * HN + h) * CN) * VP * VP;
  const float al = alpha[h];
  for (int idx = threadIdx.x; idx < CN * VP * VP; idx += blockDim.x) {
    int c = idx >> 10, r2 = idx & 1023, vv = r2 >> 5, w = r2 & 31;
    float val = 0.f;
    if (vv < VN && w < VN) {
      int pair = vv * VN + w;
      val = ffnb[h * CN + c];
      for (int r = 0; r < 8; ++r) val += ffnw[(h * CN + c) * 8 + r] * aa[r][pair];
      val = al * val + adjc[h * VN * VN + pair];
    }
    sb[idx] = val;
  }
}

// ---------------------------------------------------------------------------
// Kernel 4: heavy path.  Per (b, 8-step time tile):
//   stage P[3][32][32] -> LDS via TDM async tensor load (heavily re-read)
//   GEMM A: xp[(c,t)][w] = X @ P[h]              (WMMA, K=28, A from global/L2)
//   GEMM B: v[(o,t)][w]  = wv[h] @ xp + bv       (WMMA, K=64)
//   GEMM 3: out[o][t][v] += v . s[b,h,o]         (WMMA, K=28, s zero-padded)
// Head 2 epilogue fuses BN + inner residual + ReLU.
// All fragment addresses are linear in k; LDS ~125 KB (P 12.3 + xp 56 + v 57).
// ---------------------------------------------------------------------------
__global__ void __launch_bounds__(256)
k_attention_main(const float* __restrict__ x, const float* __restrict__ wv,
                 const float* __restrict__ bv, const float* __restrict__ Ppad,
                 const float* __restrict__ s_pad, const float* __restrict__ bn_w,
                 const float* __restrict__ bn_b, float* __restrict__ out_ws) {
  __shared__ float Pl[HN * VP * VP];             // 12.3 KB, TDM staged
  __shared__ float xp_lds[CN * XPS];             // [(c*8+t)*28+w]   56 KB
  __shared__ float v_lds[CN * XPS + XPS];        // [(o*8+t)*28+w]   56.9 KB
  const int b = blockIdx.y;
  const int t0 = blockIdx.x * TT;
  const int lane = threadIdx.x & 31;
  const int wave = threadIdx.x >> 5;
  const int col = lane & 15;
  const int half = lane >> 4;
  const float RS = rsqrtf(1.0f + 1e-5f);
  const float* xg = x + ((size_t)(b * CN) * TN + t0) * VN;

#if USE_TDM
  if (wave == 0) {
    // 1-D contiguous tensor: 3072 f32 (all three padded P matrices)
    unsigned lds_off = (unsigned)(unsigned long long)(size_t)&Pl[0];
    unsigned long long ga = (unsigned long long)(size_t)Ppad;
    v4u g0;
    g0[0] = 1u;                                   // count=1 (valid), load
    g0[1] = lds_off;                              // lds_addr (bytes)
    g0[2] = (unsigned)(ga & 0xFFFFFFFFull);       // global_addr lo
    g0[3] = (unsigned)((ga >> 32) & 0x01FFFFFFull) | (2u << 30);  // hi | type=2
    v8i g1;
    g1[0] = (int)(2u << 16);                      // data_size = 4B
    g1[1] = (int)((HN * VP * VP) << 16);          // tensor_dim0 lo16
    g1[2] = (int)(1u << 16);                      // dim0 hi=0 | tensor_dim1=1
    g1[3] = (int)((HN * VP * VP) << 16);          // dim1 hi=0 | tile_dim0
    g1[4] = 1;                                    // tile_dim1=1 | tile_dim2=0
    g1[5] = (int)(HN * VP * VP);                  // tensor_dim0_stride lo32
    g1[6] = 0;
    g1[7] = 0;
    v4i gz = {0, 0, 0, 0};
#if __clang_major__ >= 23
    v8i gz8 = {0, 0, 0, 0, 0, 0, 0, 0};
    __builtin_amdgcn_tensor_load_to_lds(g0, g1, gz, gz, gz8, 0);
#else
    __builtin_amdgcn_tensor_load_to_lds(g0, g1, gz, gz, 0);
#endif
    __builtin_amdgcn_s_wait_tensorcnt(0);
  }
#else
  for (int i = threadIdx.x; i < HN * VP * VP; i += 256) Pl[i] = Ppad[i];
#endif
  // warm L2 for this block's s slices
  for (int i = threadIdx.x; i < HN * CN; i += 256)
    __builtin_prefetch(s_pad + ((size_t)b * HN * CN + i) * (VP * VP), 0, 0);
  __syncthreads();

  for (int h = 0; h < HN; ++h) {
    const float* wvh = wv + h * CN * CN;
    const float* Phl = Pl + h * VP * VP;

    // ---- GEMM A: xp = X @ P   (32 M-tiles of rows (c,t), 2 N-tiles, K=28)
    for (int j = 0; j < 4; ++j) {
      int mt = wave * 4 + j;
      int r = mt * 16 + col;
      const float* xrow = xg + (size_t)(r >> 3) * (TN * VN) + (r & 7) * VN;
      v2f ak[7];
#pragma unroll
      for (int k = 0; k < 6; ++k) {
        int kk0 = k * 4 + 2 * half;
        ak[k].x = xrow[kk0];
        ak[k].y = xrow[kk0 + 1];
      }
      // k=6 covers u=24..27; only u=24 is real (half==0 .x), rest is K-padding
      ak[6].x = (half == 0) ? xrow[24] : 0.f;
      ak[6].y = 0.f;
#pragma unroll
      for (int nt = 0; nt < 2; ++nt) {
        int n = nt * 16 + col;  // w
        v8f acc = vzero8();
#pragma unroll
        for (int k = 0; k < 7; ++k) {
          int kk0 = k * 4 + 2 * half;
          v2f bb;
          bb.x = Phl[kk0 * VP + n];
          bb.y = Phl[(kk0 + 1) * VP + n];
          acc = wmma4(ak[k], bb, acc);
        }
        if (n < 28) {  // xp row stride is 28; w>=25 columns are zero anyway
#pragma unroll
          for (int i = 0; i < 8; ++i) {
            int m = mt * 16 + i + 8 * half;
            xp_lds[m * 28 + n] = acc[i];
          }
        }
      }
    }
    __syncthreads();

    // ---- GEMM B: v = wv @ xp + bv   (4 M-tiles, 14 N-tiles of (t,w), K=64)
    {
      int mt = wave & 3;
      v2f ak[16];
      float bvr[8];
#pragma unroll
      for (int k = 0; k < 16; ++k) {
        int kk0 = k * 4 + 2 * half;
        ak[k].x = wvh[(mt * 16 + col) * CN + kk0];
        ak[k].y = wvh[(mt * 16 + col) * CN + kk0 + 1];
      }
#pragma unroll
      for (int i = 0; i < 8; ++i) bvr[i] = bv[h * CN + mt * 16 + i + 8 * half];
      for (int j = 0; j < 7; ++j) {
        int nt = (wave >> 2) * 7 + j;
        int n = nt * 16 + col;   // t*28+w flat
        v8f acc;
#pragma unroll
        for (int i = 0; i < 8; ++i) acc[i] = bvr[i];
#pragma unroll
        for (int k = 0; k < 16; ++k) {
          int kk0 = k * 4 + 2 * half;
          v2f bb;
          bb.x = xp_lds[kk0 * XPS + n];
          bb.y = xp_lds[(kk0 + 1) * XPS + n];
          acc = wmma4(ak[k], bb, acc);
        }
#pragma unroll
        for (int i = 0; i < 8; ++i) {
          int m = mt * 16 + i + 8 * half;  // o
          v_lds[m * XPS + n] = acc[i];
        }
      }
    }
    __syncthreads();

    // ---- GEMM 3: out[o][t][vv] = sum_w v[(o,t)][w] * s[o][vv][w]
    for (int jo = 0; jo < 8; ++jo) {
      int o = wave + 8 * jo;
      v2f ak[7];
#pragma unroll
      for (int k = 0; k < 7; ++k) {
        int kk0 = k * 4 + 2 * half;
        ak[k].x = v_lds[(o * TT + col) * 28 + kk0];  // rows t>=8: discarded D rows
        ak[k].y = v_lds[(o * TT + col) * 28 + kk0 + 1];
      }
      const float* sb = s_pad + (size_t)((b * HN + h) * CN + o) * (VP * VP);
#pragma unroll
      for (int nt = 0; nt < 2; ++nt) {
        const float* sp = sb + (nt * 16 + col) * VP;  // B[w][vv] = s[vv][w]
        v8f acc = vzero8();
#pragma unroll
        for (int k = 0; k < 7; ++k) {
          int kk0 = k * 4 + 2 * half;
          v2f bb;
          bb.x = sp[kk0];
          bb.y = sp[kk0 + 1];
          acc = wmma4(ak[k], bb, acc);
        }
        if (half == 0) {
          int vv = nt * 16 + col;
          if (vv < VN) {
#pragma unroll
            for (int i = 0; i < TT; ++i) {  // valid D rows: m = i = t
              size_t idx = ((size_t)(b * CN + o) * TN + (t0 + i)) * VN + vv;
              if (h == 0) {
                out_ws[idx] = acc[i];
              } else if (h == 1) {
                out_ws[idx] += acc[i];
              } else {
                float tot = out_ws[idx] + acc[i];
                float val = tot * (bn_w[o] * RS) + bn_b[o] + x[idx];
                out_ws[idx] = fmaxf(val, 0.f);
              }
            }
          }
        }
      }
    }
    __syncthreads();
  }
}

// ---------------------------------------------------------------------------
// Kernel 5: feature extraction. Per (b, 16-step tile + halo 4), per branch:
//   Phase A: 1x1 conv C->16 (WMMA, M=16 exact) + BN1 + ReLU -> LDS
//   Phase B: temporal op + BN2 + outer residual + ReLU -> d_out
// ---------------------------------------------------------------------------
__global__ void __launch_bounds__(256)
k_feature_extract(const float* __restrict__ out_ws, const float* __restrict__ x,
                  const float* __restrict__ fe_cw, const float* __restrict__ fe_cb,
                  const float* __restrict__ fe_bn1w, const float* __restrict__ fe_bn1b,
                  const float* __restrict__ fe_bn2w, const float* __restrict__ fe_bn2b,
                  const float* __restrict__ w1, const float* __restrict__ b1,
                  const float* __restrict__ w2, const float* __restrict__ b2,
                  float* __restrict__ dout) {
  const int NSLOT = TT4 + 2 * HALO;  // 24
  __shared__ float hbuf[16 * (TT4 + 2 * HALO) * VP];  // 48 KB
  const int b = blockIdx.y;
  const int t0 = blockIdx.x * TT4;
  const int lane = threadIdx.x & 31;
  const int wave = threadIdx.x >> 5;
  const int col = lane & 15;
  const int half = lane >> 4;
  const float RS = rsqrtf(1.0f + 1e-5f);

  for (int bi = 0; bi < 4; ++bi) {
    const float* cw = fe_cw + bi * 16 * CN;
    const float padv = (bi == 2) ? -INFINITY : 0.f;
    v2f aw[16];
#pragma unroll
    for (int k = 0; k < 16; ++k) {
      int kk0 = k * 4 + 2 * half;
      aw[k].x = cw[col * CN + kk0];
      aw[k].y = cw[col * CN + kk0 + 1];
    }

    // Phase A: N = (slot,u) 24*32 -> 48 N-tiles, M = 16 (exact), K = 64
    for (int tile = wave; tile < 48; tile += 8) {
      int n = tile * 16 + col;
      int slot = n >> 5, u = n & 31;
      int tg = t0 - HALO + slot;
      bool ok = (u < VN) && (tg >= 0) && (tg < TN);
      int tgc = tg < 0 ? 0 : (tg > TN - 1 ? TN - 1 : tg);
      int uc = u < VN ? u : VN - 1;
      const float* orow = out_ws + ((size_t)(b * CN) * TN + tgc) * VN + uc;
      v8f acc = vzero8();
#pragma unroll
      for (int k = 0; k < 16; ++k) {
        int kk0 = k * 4 + 2 * half;
        float f0 = orow[(size_t)kk0 * TN * VN];   // unconditional (clamped)
        float f1 = orow[(size_t)(kk0 + 1) * TN * VN];
        v2f bb;
        bb.x = ok ? f0 : 0.f;                     // branchless select
        bb.y = ok ? f1 : 0.f;
        acc = wmma4(aw[k], bb, acc);
      }
#pragma unroll
      for (int i = 0; i < 8; ++i) {
        int o = i + 8 * half;
        float val;
        if (tg >= 0 && tg < TN) {
          float v = acc[i] + fe_cb[bi * 16 + o];
          v = v * (fe_bn1w[bi * 16 + o] * RS) + fe_bn1b[bi * 16 + o];
          val = fmaxf(v, 0.f);
        } else {
          val = padv;  // conv/avg zero-pad, max-pool -inf pad
        }
        hbuf[(o * NSLOT + slot) * VP + u] = val;
      }
    }
    __syncthreads();

    // Phase B
    for (int idx = threadIdx.x; idx < 16 * TT4 * VP; idx += blockDim.x) {
      int o = idx >> 9;
      int r2 = idx & 511;
      int tl = r2 >> 5, vv = r2 & 31;
      if (vv >= VN) continue;
      int slot = tl + HALO;
      float val;
      if (bi == 0) {
        val = b1[o];
        for (int ic = 0; ic < 16; ++ic)
          for (int d = 0; d < 5; ++d)
            val += w1[(o * 16 + ic) * 5 + d] * hbuf[(ic * NSLOT + slot + d - 2) * VP + vv];
      } else if (bi == 1) {
        val = b2[o];
        for (int ic = 0; ic < 16; ++ic)
          for (int d = 0; d < 5; ++d)
            val += w2[(o * 16 + ic) * 5 + d] * hbuf[(ic * NSLOT + slot + 2 * d - 4) * VP + vv];
      } else if (bi == 2) {
        val = fmaxf(fmaxf(hbuf[(o * NSLOT + slot - 1) * VP + vv],
                          hbuf[(o * NSLOT + slot) * VP + vv]),
                    hbuf[(o * NSLOT + slot + 1) * VP + vv]);
      } else {
        val = (hbuf[(o * NSLOT + slot - 1) * VP + vv] +
               hbuf[(o * NSLOT + slot) * VP + vv] +
               hbuf[(o * NSLOT + slot + 1) * VP + vv]) * (1.0f / 3.0f);
      }
      val = val * (fe_bn2w[bi * 16 + o] * RS) + fe_bn2b[bi * 16 + o];
      int oc = bi * 16 + o;
      size_t gi = ((size_t)(b * CN + oc) * TN + (t0 + tl)) * VN + vv;
      dout[gi] = fmaxf(val + x[gi], 0.f);
    }
    __syncthreads();
  }
}

// ---------------------------------------------------------------------------
extern "C" void kernel_launch(void* const* d_in, const int* in_sizes, int n_in,
                              void* d_out, int out_size, void* d_ws, size_t ws_size,
                              hipStream_t stream) {
  (void)in_sizes; (void)n_in; (void)out_size; (void)ws_size;
  const float* x         = (const float*)d_in[0];
  const float* adjc      = (const float*)d_in[1];
  const float* log_alpha = (const float*)d_in[2];
  const float* wk   = (const float*)d_in[3];
  const float* bk   = (const float*)d_in[4];
  const float* wq   = (const float*)d_in[5];
  const float* bq   = (const float*)d_in[6];
  const float* wv   = (const float*)d_in[7];
  const float* bv   = (const float*)d_in[8];
  const float* dpw1 = (const float*)d_in[9];
  const float* dpb1 = (const float*)d_in[10];
  const float* dpw2 = (const float*)d_in[11];
  const float* dpb2 = (const float*)d_in[12];
  const float* ffnw = (const float*)d_in[13];
  const float* ffnb = (const float*)d_in[14];
  const float* alpha = (const float*)d_in[15];
  const float* bn_w  = (const float*)d_in[16];
  const float* bn_b  = (const float*)d_in[17];
  const float* fe_cw   = (const float*)d_in[18];
  const float* fe_cb   = (const float*)d_in[19];
  const float* fe_bn1w = (const float*)d_in[20];
  const float* fe_bn1b = (const float*)d_in[21];
  const float* fe_bn2w = (const float*)d_in[22];
  const float* fe_bn2b = (const float*)d_in[23];
  const float* w1 = (const float*)d_in[24];
  const float* b1 = (const float*)d_in[25];
  const float* w2 = (const float*)d_in[26];
  const float* b2 = (const float*)d_in[27];

  // workspace layout (floats): P | xbar | s_pad | out_ws   (~156 MB total)
  float* wsf   = (float*)d_ws;
  float* Ppad  = wsf;                                      // 3*32*32
  float* xbar  = Ppad + HN * VP * VP;                      // 64*64*32
  float* s_pad = xbar + BB * CN * VP;                      // 64*3*64*32*32
  float* out_ws = s_pad + (size_t)BB * HN * CN * VP * VP;  // 64*64*256*25

  k_sinkhorn<<<1, 32, 0, stream>>>(log_alpha, Ppad);
  k_xbar<<<(BB * CN * VP + 255) / 256, 256, 0, stream>>>(x, xbar);
  dim3 gs(HN, BB);
  k_attn_s<<<gs, 256, 0, stream>>>(xbar, Ppad, wk, bk, wq, bq, dpw1, dpb1, dpw2,
                                   dpb2, ffnw, ffnb, alpha, adjc, s_pad);
  dim3 gm(TN / TT, BB);
  k_attention_main<<<gm, 256, 0, stream>>>(x, wv, bv, Ppad, s_pad, bn_w, bn_b,
                                           out_ws);
  dim3 gf(TN / TT4, BB);
  k_feature_extract<<<gf, 256, 0, stream>>>(out_ws, x, fe_cw, fe_cb, fe_bn1w,
                                            fe_bn1b, fe_bn2w, fe_bn2b, w1, b1,
                                            w2, b2, (float*)d_out);
}